// Local_match_61546881352466
// MI455X (gfx1250) — compile-verified
//
#include <hip/hip_runtime.h>
#include <hip/hip_bf16.h>

#define BATCH 64
#define CCH   256
#define SPIX  4096          // 64*64
#define KCLS  5
#define PLOC  3
#define NPROTO 15
#define NPAD   16
#define NCG   16            // channel groups for norm pass
#define CPG   (CCH / NCG)   // 16 channels per group

typedef __attribute__((ext_vector_type(2))) float v2f;
typedef __attribute__((ext_vector_type(8))) float v8f;

// ---------------- 1) normalize prototypes -> snT[c][n] (n=15 valid, col 15 = 0)
__global__ void k_sfc(const float* __restrict__ SFC, float* __restrict__ snT) {
    __shared__ float rn[NPROTO];
    int tid = threadIdx.x;
    if (tid < NPROTO) {
        float s = 0.f;
        for (int c = 0; c < CCH; ++c) { float v = SFC[tid * CCH + c]; s += v * v; }
        rn[tid] = 1.0f / fmaxf(sqrtf(s), 1e-12f);
    }
    __syncthreads();
    for (int idx = tid; idx < CCH * NPAD; idx += 256) {
        int c = idx >> 4, kp = idx & 15;
        snT[idx] = (kp < NPROTO) ? SFC[kp * CCH + c] * rn[kp] : 0.f;
    }
}

// ---------------- 2) CCL per batch (min-index propagation in LDS) + counts
__global__ void __launch_bounds__(256) k_ccl(const int* __restrict__ mask,
                                             int* __restrict__ labG,
                                             int* __restrict__ cntG) {
    __shared__ int lab[SPIX];
    __shared__ int cnt[SPIX];
    __shared__ int sChanged;
    int b = blockIdx.x, tid = threadIdx.x;
    const int* m = mask + b * SPIX;
    for (int p = tid; p < SPIX; p += 256) lab[p] = (m[p] > 0) ? p : SPIX;

    const int PPT = SPIX / 256;  // 16
    int p0 = tid * PPT;
    while (true) {
        __syncthreads();
        if (tid == 0) sChanged = 0;
        __syncthreads();
        bool my = false;
        for (int j = 0; j < PPT; ++j) {
            int p = p0 + j;
            int l = lab[p];
            if (l == SPIX) continue;
            int y = p >> 6, x = p & 63;
            int mn = l;
            if (y > 0)  { int q = p - 64; mn = min(mn, lab[q]);
                          if (x > 0)  mn = min(mn, lab[q - 1]);
                          if (x < 63) mn = min(mn, lab[q + 1]); }
            if (x > 0)  mn = min(mn, lab[p - 1]);
            if (x < 63) mn = min(mn, lab[p + 1]);
            if (y < 63) { int q = p + 64; mn = min(mn, lab[q]);
                          if (x > 0)  mn = min(mn, lab[q - 1]);
                          if (x < 63) mn = min(mn, lab[q + 1]); }
            if (mn < l) { lab[p] = mn; my = true; }
        }
        if (my) sChanged = 1;
        __syncthreads();
        if (sChanged == 0) break;
    }

    for (int p = tid; p < SPIX; p += 256) cnt[p] = 0;
    __syncthreads();
    {   // run-length compressed counting
        int cur = -1, acc = 0;
        for (int j = 0; j < PPT; ++j) {
            int l = lab[p0 + j];
            if (l == cur) { acc++; }
            else {
                if (cur >= 0 && cur < SPIX) atomicAdd(&cnt[cur], acc);
                cur = l; acc = 1;
            }
        }
        if (cur >= 0 && cur < SPIX) atomicAdd(&cnt[cur], acc);
    }
    __syncthreads();
    for (int p = tid; p < SPIX; p += 256) {
        labG[b * SPIX + p] = lab[p];
        cntG[b * SPIX + p] = cnt[p];
    }
}

// ---------------- 3) zero scratch accumulators (normsqG + compProj, contiguous)
__global__ void k_zero(float* __restrict__ p, long n) {
    long i = (long)blockIdx.x * blockDim.x + threadIdx.x;
    if (i < n) p[i] = 0.f;
}

// ---------------- 4) projection GEMM via fp32 WMMA + run-length atomic segment-sum
__global__ void __launch_bounds__(256) k_proj(const float* __restrict__ feat,
                                              const float* __restrict__ snT,
                                              const int* __restrict__ labG,
                                              float* __restrict__ compProj) {
    __shared__ float sB[CCH * NPAD];   // 16 KB, B matrix [K=256][N=16]
    int tid = threadIdx.x;
    for (int i = tid; i < CCH * NPAD; i += 256) sB[i] = snT[i];
    __syncthreads();

    int wave = tid >> 5;
    int lane = tid & 31;
    int T = blockIdx.x * 8 + wave;       // one 16-pixel row-tile per wave
    int gp0 = T * 16;                    // global pixel index (b*S + p)
    int b = gp0 >> 12;
    int pin = gp0 & (SPIX - 1);
    const float* fb = feat + (size_t)b * CCH * SPIX;

    int r  = lane & 15;                  // A: row M ; B: col N
    int kh = (lane >> 4) << 1;           // K sub-offset 0 / 2 per half-wave

    v8f acc = {};
    for (int k0 = 0; k0 < CCH; k0 += 4) {
        int c0 = k0 + kh;
        v2f a, bb;
        a.x  = fb[(size_t)c0 * SPIX + pin + r];
        a.y  = fb[(size_t)(c0 + 1) * SPIX + pin + r];
        bb.x = sB[c0 * NPAD + r];
        bb.y = sB[(c0 + 1) * NPAD + r];
        acc = __builtin_amdgcn_wmma_f32_16x16x4_f32(
            false, a, false, bb, (short)0, acc, false, false);
    }

    // C/D layout: lane holds col r, rows rowHi+0..7 (consecutive pixels)
    int col   = r;
    int rowHi = (lane >> 4) << 3;        // 0 or 8
    if (col < NPROTO) {
        const int* lb = labG + gp0;
        int cur = -1; float s = 0.f;
        #pragma unroll
        for (int i = 0; i < 8; ++i) {
            int l = lb[rowHi + i];
            float v = acc[i];
            if (l == cur) { s += v; }
            else {
                if (cur >= 0 && cur < SPIX)
                    atomicAdd(&compProj[(((size_t)b << 12) + cur) * NPAD + col], s);
                cur = l; s = v;
            }
        }
        if (cur >= 0 && cur < SPIX)
            atomicAdd(&compProj[(((size_t)b << 12) + cur) * NPAD + col], s);
    }
}

// ---------------- 5) ||sum_comp||^2, parallel over (batch, channel group)
__global__ void __launch_bounds__(256) k_norm(const float* __restrict__ feat,
                                              const int* __restrict__ labG,
                                              float* __restrict__ normsqG) {
    __shared__ int   lab[SPIX];        // 16 KB
    __shared__ float compSum[SPIX];    // 16 KB
    __shared__ float nsq[SPIX];        // 16 KB
    int bid = blockIdx.x, tid = threadIdx.x;
    int b  = bid / NCG;
    int cg = bid % NCG;
    for (int p = tid; p < SPIX; p += 256) { lab[p] = labG[b * SPIX + p]; nsq[p] = 0.f; }
    __syncthreads();
    const float* fb = feat + (size_t)b * CCH * SPIX;
    int p0 = tid * 16;
    for (int cc = 0; cc < CPG; ++cc) {
        int c = cg * CPG + cc;
        for (int p = tid; p < SPIX; p += 256) compSum[p] = 0.f;
        __syncthreads();
        const float* fc = fb + (size_t)c * SPIX;
        int cur = -1; float s = 0.f;
        for (int j = 0; j < 16; ++j) {
            int p = p0 + j;
            int l = lab[p];
            float v = fc[p];
            if (l == cur) { s += v; }
            else {
                if (cur >= 0 && cur < SPIX) atomicAdd(&compSum[cur], s);
                cur = l; s = v;
            }
        }
        if (cur >= 0 && cur < SPIX) atomicAdd(&compSum[cur], s);
        __syncthreads();
        for (int p = tid; p < SPIX; p += 256) { float t = compSum[p]; nsq[p] += t * t; }
        __syncthreads();
    }
    // sparse flush: only component-root slots are nonzero
    for (int p = tid; p < SPIX; p += 256) {
        float v = nsq[p];
        if (v != 0.f) atomicAdd(&normsqG[b * SPIX + p], v);
    }
}

// ---------------- 6) final score: max over P, mean over valid components
__global__ void __launch_bounds__(256) k_score(const float* __restrict__ compProj,
                                               const float* __restrict__ normsqG,
                                               const int* __restrict__ cntG,
                                               float* __restrict__ out) {
    __shared__ float sAcc[KCLS];
    __shared__ int sNv;
    int b = blockIdx.x, tid = threadIdx.x;
    if (tid == 0) sNv = 0;
    if (tid < KCLS) sAcc[tid] = 0.f;
    __syncthreads();
    float acc[KCLS] = {0.f, 0.f, 0.f, 0.f, 0.f};
    int nv = 0;
    for (int s0 = tid; s0 < SPIX; s0 += 256) {
        if (cntG[b * SPIX + s0] > 0) {
            nv++;
            float ns  = normsqG[b * SPIX + s0];
            float inv = (ns > 0.f) ? rsqrtf(ns) : 0.f;
            const float* pr = compProj + (((size_t)b << 12) + s0) * NPAD;
            for (int k = 0; k < KCLS; ++k) {
                float mx = -1e30f;
                for (int pp = 0; pp < PLOC; ++pp)
                    mx = fmaxf(mx, pr[k * PLOC + pp] * inv);
                acc[k] += mx;
            }
        }
    }
    for (int k = 0; k < KCLS; ++k) atomicAdd(&sAcc[k], acc[k]);
    atomicAdd(&sNv, nv);
    __syncthreads();
    if (tid < KCLS) out[b * KCLS + tid] = sAcc[tid] / (float)max(sNv, 1);
}

extern "C" void kernel_launch(void* const* d_in, const int* in_sizes, int n_in,
                              void* d_out, int out_size, void* d_ws, size_t ws_size,
                              hipStream_t stream) {
    const float* feat = (const float*)d_in[0];   // [64,256,64,64] f32
    const int*   mask = (const int*)d_in[1];     // [64,64,64] i32
    const float* SFC  = (const float*)d_in[2];   // [5,3,256] f32
    float* out = (float*)d_out;                  // [64,5] f32

    char* ws = (char*)d_ws;
    size_t off = 0;
    float* snT      = (float*)(ws + off); off += (size_t)CCH * NPAD * sizeof(float);
    int*   labG     = (int*)  (ws + off); off += (size_t)BATCH * SPIX * sizeof(int);
    int*   cntG     = (int*)  (ws + off); off += (size_t)BATCH * SPIX * sizeof(int);
    float* normsqG  = (float*)(ws + off); off += (size_t)BATCH * SPIX * sizeof(float);
    float* compProj = (float*)(ws + off); off += (size_t)BATCH * SPIX * NPAD * sizeof(float);
    (void)ws_size; (void)in_sizes; (void)n_in; (void)out_size;

    k_sfc<<<1, 256, 0, stream>>>(SFC, snT);
    k_ccl<<<BATCH, 256, 0, stream>>>(mask, labG, cntG);
    // normsqG and compProj are contiguous: zero both in one launch
    long nz = (long)BATCH * SPIX * (1 + NPAD);
    k_zero<<<(int)((nz + 255) / 256), 256, 0, stream>>>(normsqG, nz);
    k_proj<<<BATCH * SPIX / 128, 256, 0, stream>>>(feat, snT, labG, compProj);
    k_norm<<<BATCH * NCG, 256, 0, stream>>>(feat, labG, normsqG);
    k_score<<<BATCH, 256, 0, stream>>>(compProj, normsqG, cntG, out);
}